// ColdPrompt_56805237457152
// MI455X (gfx1250) — compile-verified
//
#include <hip/hip_runtime.h>
#include <hip/hip_bf16.h>

typedef __attribute__((ext_vector_type(2))) float v2f;
typedef __attribute__((ext_vector_type(4))) float v4f;
typedef __attribute__((ext_vector_type(8))) float v8f;

#define B_SZ  256
#define P_SZ  4
#define D_SZ  768
#define U_SZ  32
#define IN_SZ (P_SZ * D_SZ)      // 3072

#define MBLK  64
#define NBLK  128
#define KC    32
#define NK    (IN_SZ / KC)       // 96
#define A_PAD 36                 // 32 + 4: 16B-aligned rows, bank-spread frag reads

__global__ __launch_bounds__(256)
void coldprompt_user_gemm(const float* __restrict__ weight,  // [B, IN] (== [B,P,D] flat)
                          const float* __restrict__ Wspec,   // [U, IN, D]
                          const float* __restrict__ bspec,   // [U, D]
                          float* __restrict__ out)           // [U*B, D]
{
    __shared__ float As[2][MBLK][A_PAD];        // 2*64*36*4  = 18 KB
    __shared__ float Bs[2][KC / 2][NBLK][2];    // 2*16*128*2*4 = 32 KB, K-pair interleaved

    const int u  = blockIdx.z;
    const int m0 = blockIdx.y * MBLK;
    const int n0 = blockIdx.x * NBLK;

    const int tid  = threadIdx.x;
    const int lane = tid & 31;
    const int wave = tid >> 5;             // 0..7
    const int wm   = (wave & 1) * 32;      // wave M offset within block tile
    const int wn   = (wave >> 1) * 32;     // wave N offset within block tile

    const float* Ag = weight + (size_t)m0 * IN_SZ;
    const float* Bg = Wspec + (size_t)u * IN_SZ * D_SZ + n0;

    // A tile loader: 64x32 floats -> two float4 per thread (row ar, cols ac..ac+7)
    const int ar = tid >> 2;               // 0..63
    const int ac = (tid & 3) * 8;          // 0,8,16,24
    // B tile loader: 32x128 floats as 16 row-pairs; thread owns (row pair brp, 8 cols)
    const int brp = tid >> 4;              // 0..15
    const int bcc = (tid & 15) * 8;        // 0..120

    v4f a0r, a1r;                          // A staging
    v4f b00, b01, b10, b11;                // B staging: rows 2*brp and 2*brp+1

    auto loadGlobal = [&](int kt) {
        const int k0 = kt * KC;
        const float* ap = Ag + (size_t)ar * IN_SZ + k0 + ac;
        a0r = *(const v4f*)(ap);
        a1r = *(const v4f*)(ap + 4);
        const float* bp = Bg + (size_t)(k0 + 2 * brp) * D_SZ + bcc;
        b00 = *(const v4f*)(bp);
        b01 = *(const v4f*)(bp + 4);
        b10 = *(const v4f*)(bp + D_SZ);
        b11 = *(const v4f*)(bp + D_SZ + 4);
    };
    auto storeLds = [&](int buf) {
        *(v4f*)&As[buf][ar][ac]     = a0r;
        *(v4f*)&As[buf][ar][ac + 4] = a1r;
        // interleave K-pair so a B frag {B[k][n],B[k+1][n]} is one contiguous b64
        v4f i0 = {b00.x, b10.x, b00.y, b10.y};
        v4f i1 = {b00.z, b10.z, b00.w, b10.w};
        v4f i2 = {b01.x, b11.x, b01.y, b11.y};
        v4f i3 = {b01.z, b11.z, b01.w, b11.w};
        float* q = &Bs[buf][brp][bcc][0];
        *(v4f*)(q)      = i0;
        *(v4f*)(q + 4)  = i1;
        *(v4f*)(q + 8)  = i2;
        *(v4f*)(q + 12) = i3;
    };

    v8f acc00 = {}, acc01 = {}, acc10 = {}, acc11 = {};

    // WMMA f32 16x16x4 fragment indexing (wave32):
    //  A: lane m = l&15; VGPR0/1 hold K = 2*(l>>4) + {0,1}
    //  B: lane n = l&15; VGPR0/1 hold K = 2*(l>>4) + {0,1}
    //  C/D: VGPR i -> row 8*(l>>4)+i, col l&15
    const int mrow = lane & 15;
    const int ncol = lane & 15;
    const int kb   = (lane >> 4) * 2;      // K offset within 4-chunk
    const int kh   = (lane >> 4);          // K-pair offset within 4-chunk

    loadGlobal(0);
    storeLds(0);

    for (int kt = 0; kt < NK; ++kt) {
        __syncthreads();
        const int buf = kt & 1;
        if (kt + 1 < NK) loadGlobal(kt + 1);   // prefetch next chunk into regs

        #pragma unroll
        for (int k4 = 0; k4 < KC; k4 += 4) {
            v2f a0 = *(const v2f*)&As[buf][wm + mrow][k4 + kb];
            v2f a1 = *(const v2f*)&As[buf][wm + 16 + mrow][k4 + kb];
            const int k2 = (k4 >> 1) + kh;
            v2f b0 = *(const v2f*)&Bs[buf][k2][wn + ncol][0];
            v2f b1 = *(const v2f*)&Bs[buf][k2][wn + 16 + ncol][0];
            acc00 = __builtin_amdgcn_wmma_f32_16x16x4_f32(false, a0, false, b0, (short)0, acc00, false, false);
            acc01 = __builtin_amdgcn_wmma_f32_16x16x4_f32(false, a0, false, b1, (short)0, acc01, false, false);
            acc10 = __builtin_amdgcn_wmma_f32_16x16x4_f32(false, a1, false, b0, (short)0, acc10, false, false);
            acc11 = __builtin_amdgcn_wmma_f32_16x16x4_f32(false, a1, false, b1, (short)0, acc11, false, false);
        }

        if (kt + 1 < NK) {
            __syncthreads();                   // prior readers of next buf are done
            storeLds((kt + 1) & 1);
        }
    }

    // Epilogue: add bias (depends only on n), store 2x2 tiles
    const float bias0 = bspec[u * D_SZ + n0 + wn + ncol];
    const float bias1 = bspec[u * D_SZ + n0 + wn + 16 + ncol];
    const int rbase = (lane >> 4) * 8;
    float* outBase = out + ((size_t)u * B_SZ + m0 + wm) * D_SZ + n0 + wn + ncol;
    #pragma unroll
    for (int i = 0; i < 8; ++i) {
        const int r0 = rbase + i;
        outBase[(size_t)r0 * D_SZ]              = acc00[i] + bias0;
        outBase[(size_t)r0 * D_SZ + 16]         = acc01[i] + bias1;
        outBase[(size_t)(r0 + 16) * D_SZ]       = acc10[i] + bias0;
        outBase[(size_t)(r0 + 16) * D_SZ + 16]  = acc11[i] + bias1;
    }
}

__global__ __launch_bounds__(256)
void coldprompt_mean(const float* __restrict__ weight,  // [B, P, D]
                     float* __restrict__ out)           // [B, D]
{
    const int idx = blockIdx.x * 256 + threadIdx.x;
    if (idx >= B_SZ * D_SZ) return;
    const int b = idx / D_SZ;
    const int d = idx - b * D_SZ;
    const float* p = weight + (size_t)b * IN_SZ + d;
    out[idx] = 0.25f * (p[0] + p[D_SZ] + p[2 * D_SZ] + p[3 * D_SZ]);
}

extern "C" void kernel_launch(void* const* d_in, const int* in_sizes, int n_in,
                              void* d_out, int out_size, void* d_ws, size_t ws_size,
                              hipStream_t stream) {
    const float* weight = (const float*)d_in[0];  // [B,P,D] fp32
    const float* Wspec  = (const float*)d_in[1];  // [U,IN,D] fp32
    const float* bspec  = (const float*)d_in[2];  // [U,D] fp32
    float* out = (float*)d_out;                   // [U*B*D] then [B*D]

    dim3 grid(D_SZ / NBLK, B_SZ / MBLK, U_SZ);    // (6, 4, 32)
    coldprompt_user_gemm<<<grid, 256, 0, stream>>>(weight, Wspec, bspec, out);

    float* meanOut = out + (size_t)U_SZ * B_SZ * D_SZ;
    coldprompt_mean<<<(B_SZ * D_SZ + 255) / 256, 256, 0, stream>>>(weight, meanOut);
}